// Twist2Mat_48696339202533
// MI455X (gfx1250) — compile-verified
//
#include <hip/hip_runtime.h>
#include <cstdint>

typedef float v4f __attribute__((ext_vector_type(4)));

#define TILE_ROT   1024                 // rotations per tile
#define TILE_BYTES (TILE_ROT * 12)      // 12288 B input per tile
#define WAVE_ROT   128                  // rotations per wave per tile
#define WAVE_BYTES (WAVE_ROT * 12)      // 1536 B per wave per tile
#define NBUF       2

// ---- CDNA5 async global->LDS copy (ASYNCcnt path) -------------------------
__device__ __forceinline__ void async_b128(unsigned ldsAddr, unsigned gOff, const void* base) {
    // vdst = LDS byte address VGPR, vaddr = 32-bit offset, saddr = 64-bit base (GVS mode)
    asm volatile("global_load_async_to_lds_b128 %0, %1, %2"
                 :: "v"(ldsAddr), "v"(gOff), "s"(base) : "memory");
}
__device__ __forceinline__ void wait_async_le3() {
    asm volatile("s_wait_asynccnt 0x3" ::: "memory");
}
__device__ __forceinline__ void wait_async_0() {
    asm volatile("s_wait_asynccnt 0x0" ::: "memory");
}

// ---- Rodrigues closed form ------------------------------------------------
// axis = w / max(||w||,1e-5);  R = I + s*A + (1-c)*(a a^T - ||a||^2 I)
__device__ __forceinline__ void rodrigues(float x, float y, float z, float* r) {
    float n2  = x * x + y * y + z * z;
    float th  = fmaxf(__builtin_sqrtf(n2), 1e-5f);
    float inv = 1.0f / th;
    float a0 = x * inv, a1 = y * inv, a2 = z * inv;
    float s, c;
    __sincosf(th, &s, &c);
    float c1   = 1.0f - c;
    float m2   = n2 * inv * inv;          // ||axis||^2 (1 unless clamped)
    float diag = 1.0f - c1 * m2;
    float p01 = c1 * a0 * a1, p02 = c1 * a0 * a2, p12 = c1 * a1 * a2;
    float s0 = s * a0, s1 = s * a1, s2 = s * a2;
    r[0] = diag + c1 * a0 * a0;  r[1] = p01 - s2;             r[2] = p02 + s1;
    r[3] = p01 + s2;             r[4] = diag + c1 * a1 * a1;  r[5] = p12 - s0;
    r[6] = p02 - s1;             r[7] = p12 + s0;             r[8] = diag + c1 * a2 * a2;
}

// ---- main streaming kernel: async double-buffered LDS staging -------------
__global__ __launch_bounds__(256) void twist2mat_async(const float* __restrict__ in,
                                                       float* __restrict__ out,
                                                       int nTiles) {
    __shared__ __align__(16) unsigned char smem[NBUF * TILE_BYTES];
    const unsigned ldsBase = (unsigned)(uintptr_t)(void*)smem;   // low 32b = LDS offset
    const int wave = threadIdx.x >> 5;
    const int lane = threadIdx.x & 31;
    const unsigned waveLds = (unsigned)wave * WAVE_BYTES;

    auto issue = [&](int tile, int buf) {
        unsigned g = (unsigned)tile * TILE_BYTES + waveLds + (unsigned)lane * 16u;
        unsigned l = ldsBase + (unsigned)buf * TILE_BYTES + waveLds + (unsigned)lane * 16u;
        async_b128(l,         g,         in);
        async_b128(l +  512u, g +  512u, in);
        async_b128(l + 1024u, g + 1024u, in);
    };

    const int stride = gridDim.x;
    int t0 = blockIdx.x;
    if (t0 >= nTiles) return;

    issue(t0, 0);
    int buf = 0;
    for (int t = t0; t < nTiles; t += stride) {
        const int  tn   = t + stride;
        const bool more = tn < nTiles;
        if (more) { issue(tn, buf ^ 1); wait_async_le3(); }
        else      { wait_async_0(); }

        // this lane's 4 rotations: 48 B from LDS (16B-aligned)
        const v4f* src = (const v4f*)(smem + buf * TILE_BYTES + waveLds + (unsigned)lane * 48u);
        v4f q0 = src[0], q1 = src[1], q2 = src[2];

        alignas(16) float rr[36];
        rodrigues(q0.x, q0.y, q0.z, rr + 0);
        rodrigues(q0.w, q1.x, q1.y, rr + 9);
        rodrigues(q1.z, q1.w, q2.x, rr + 18);
        rodrigues(q2.y, q2.z, q2.w, rr + 27);

        const long rotBase = (long)t * TILE_ROT + wave * WAVE_ROT + lane * 4;
        v4f* ov = (v4f*)(out + rotBase * 9);          // 144B-aligned, contiguous per lane
        const v4f* rv = (const v4f*)rr;
#pragma unroll
        for (int i = 0; i < 9; ++i)
            __builtin_nontemporal_store(rv[i], ov + i);   // output > L2 (192MB): stream it

        buf ^= 1;
    }
}

// ---- scalar tail (only if nrot % 1024 != 0; not hit for 4096x2048) --------
__global__ void twist2mat_tail(const float* __restrict__ in, float* __restrict__ out,
                               long start, long n) {
    long i = start + (long)blockIdx.x * blockDim.x + threadIdx.x;
    if (i >= n) return;
    float x = in[i * 3 + 0], y = in[i * 3 + 1], z = in[i * 3 + 2];
    float r[9];
    rodrigues(x, y, z, r);
#pragma unroll
    for (int k = 0; k < 9; ++k) out[i * 9 + k] = r[k];
}

extern "C" void kernel_launch(void* const* d_in, const int* in_sizes, int n_in,
                              void* d_out, int out_size, void* d_ws, size_t ws_size,
                              hipStream_t stream) {
    const float* in  = (const float*)d_in[0];
    float*       out = (float*)d_out;
    const long nrot   = (long)in_sizes[0] / 3;      // 8,388,608 for 4096x2048
    const long nTiles = nrot / TILE_ROT;            // 8192

    if (nTiles > 0) {
        int grid = (int)(nTiles < 1024 ? nTiles : 1024);   // 8 tiles/block pipeline depth
        twist2mat_async<<<grid, 256, 0, stream>>>(in, out, (int)nTiles);
    }
    const long done = nTiles * TILE_ROT;
    const long rem  = nrot - done;
    if (rem > 0) {
        int blocks = (int)((rem + 255) / 256);
        twist2mat_tail<<<blocks, 256, 0, stream>>>(in, out, done, nrot);
    }
}